// Net_20813411517130
// MI455X (gfx1250) — compile-verified
//
#include <hip/hip_runtime.h>
#include <hip/hip_bf16.h>
#include <math.h>

typedef __attribute__((ext_vector_type(2))) float v2f;
typedef __attribute__((ext_vector_type(8))) float v8f;

#define NKS 16

// ---------------------------------------------------------------------------
// Kernel A (one block per image, 8 waves):
//   z   = x - sum_k delta_init[k]                       (LDS, float4 loads)
//   a1  = relu(maxpool2(conv5x5(z, W1)))                (LDS, 10x12x12)
//   Aim = im2col(a1)  [64 x 252], zero-padded           (LDS)
//   Btp = W2^T pair-interleaved [126][32][2], padded    (LDS)
//   conv2 GEMM 64x252 @ 252x32 with V_WMMA_F32_16X16X4_F32, guard-free loop
//   a2  = relu(maxpool2(conv2)) -> global [20,4,4]=320
// ---------------------------------------------------------------------------
__global__ __launch_bounds__(256)
void fused_conv_kernel(const float* __restrict__ x,
                       const float* __restrict__ dinit,
                       const float* __restrict__ W1,
                       const float* __restrict__ W2,
                       float* __restrict__ a2)
{
    __shared__ float zl[784];           // 28x28 decrypted input
    __shared__ float w1l[256];          // W1 [10][1][5][5]
    __shared__ float a1l[1440];         // [10][12][12]
    __shared__ float Aim[64 * 252];     // im2col A, K padded 250->252 (zeros)
    __shared__ float Btp[126 * 64];     // B pairs: [(k>>1)][o(0..31)][k&1]

    const int img = blockIdx.x;
    const int tid = threadIdx.x;

    // ---- stage 0: z = x - sum_k delta_init[k]  (b128 loads; the only heavy HBM)
    {
        const float4* x4 = (const float4*)x;
        const float4* d4 = (const float4*)dinit;
        for (int i = tid; i < 196; i += 256) {
            float4 v = x4[img * 196 + i];
            #pragma unroll
            for (int k = 0; k < NKS; ++k) {
                const float4 d = d4[(size_t)k * 200704 + img * 196 + i];
                v.x -= d.x; v.y -= d.y; v.z -= d.z; v.w -= d.w;
            }
            ((float4*)zl)[i] = v;
        }
    }
    for (int i = tid; i < 250; i += 256) w1l[i] = W1[i];
    // Btp[(k>>1)*64 + o*2 + (k&1)] = W2[o][k], zero-padded (k>=250 or o>=20)
    for (int i = tid; i < 126 * 64; i += 256) {
        const int kk2 = i >> 6;
        const int rem = i & 63;
        const int o   = rem >> 1;
        const int k   = (kk2 << 1) | (rem & 1);
        Btp[i] = (k < 250 && o < 20) ? W2[o * 250 + k] : 0.f;
    }
    __syncthreads();

    // ---- stage 1: conv1 (1->10, 5x5) + 2x2 maxpool + relu
    for (int o = tid; o < 1440; o += 256) {
        const int oc  = o / 144;
        const int pos = o % 144;
        const int py  = pos / 12, px = pos % 12;
        const float* w = &w1l[oc * 25];
        float m = -1e30f;
        #pragma unroll
        for (int dy = 0; dy < 2; ++dy)
        #pragma unroll
        for (int dx = 0; dx < 2; ++dx) {
            const int y = 2 * py + dy, xx = 2 * px + dx;
            float acc = 0.f;
            #pragma unroll
            for (int r = 0; r < 5; ++r)
            #pragma unroll
            for (int c = 0; c < 5; ++c)
                acc += zl[(y + r) * 28 + (xx + c)] * w[r * 5 + c];
            m = fmaxf(m, acc);
        }
        a1l[o] = fmaxf(m, 0.f);
    }
    __syncthreads();

    // ---- stage 1.5: materialize im2col A (index decode done ONCE per element)
    for (int i = tid; i < 64 * 252; i += 256) {
        const int m = i / 252;
        const int k = i - m * 252;
        float v = 0.f;
        if (k < 250) {
            const int ic = k / 25;
            const int rm = k % 25;
            v = a1l[ic * 144 + ((m >> 3) + rm / 5) * 12 + ((m & 7) + rm % 5)];
        }
        Aim[i] = v;
    }
    __syncthreads();

    // ---- stage 2: guard-free WMMA GEMM, one 16x16 C tile per wave
    const int wave  = tid >> 5;
    const int lane  = tid & 31;
    const int mTile = wave & 3;          // output rows 16*mTile..+15 (oy pair)
    const int nTile = wave >> 2;         // output cols 16*nTile..+15
    const int lh    = lane & 15;
    const int kOff  = (lane >> 4) << 1;  // lanes 0-15: K+{0,1}; 16-31: K+{2,3}
    const int arow  = (mTile << 4) + lh; // im2col row (output position)
    const int ocol  = (nTile << 4) + lh; // output channel

    const float* Ap = &Aim[arow * 252 + kOff];
    const float* Bp = &Btp[(kOff >> 1) * 64 + ocol * 2];

    v8f c = {};
    for (int ks = 0; ks < 63; ++ks) {
        const v2f a = *(const v2f*)(Ap + ks * 4);    // ds_load_b64
        const v2f b = *(const v2f*)(Bp + ks * 128);  // ds_load_b64
        c = __builtin_amdgcn_wmma_f32_16x16x4_f32(false, a, false, b,
                                                  (short)0, c, false, false);
    }

    // ---- stage 3: 2x2 maxpool + relu directly on the C fragment.
    // vgpr r -> row m0+r (lanes 0-15) / m0+r+8 (lanes 16-31); pool px=i uses
    // rows {2i,2i+1} of both halves -> fmax pair + cross-half shfl.
    #pragma unroll
    for (int i = 0; i < 4; ++i) {
        float t = fmaxf(c[2 * i], c[2 * i + 1]);
        t = fmaxf(t, __shfl_xor(t, 16, 32));
        t = fmaxf(t, 0.f);
        if (lane < 16 && ocol < 20)       // pooled (py=mTile, px=i)
            a2[img * 320 + ocol * 16 + mTile * 4 + i] = t;
    }
}

// ---------------------------------------------------------------------------
// Kernel B: logits = a2 @ Wfc^T via WMMA (M=1024, K=320, N=10 padded to 16),
// Wfc zero-padded to 16 rows in LDS -> guard-free K-loop (b64 loads only),
// then relu + masked softmax on the C fragment + sum_k delta_6.
// ---------------------------------------------------------------------------
__global__ __launch_bounds__(256)
void fc_softmax_kernel(const float* __restrict__ a2,
                       const float* __restrict__ Wfc,
                       const float* __restrict__ d6,
                       float* __restrict__ out)
{
    __shared__ float wl[16 * 320];           // Wfc padded [16][320]
    const int tid = threadIdx.x;
    for (int i = tid; i < 16 * 320; i += 256)
        wl[i] = (i < 3200) ? Wfc[i] : 0.f;
    __syncthreads();

    const int wave = tid >> 5;
    const int lane = tid & 31;
    const int m0   = (blockIdx.x * 8 + wave) << 4;   // 16-image tile base
    const int lh   = lane & 15;
    const int kOff = (lane >> 4) << 1;
    const int arow = m0 + lh;
    const int o    = lh;                     // class index for this lane

    const float* ap = a2 + (size_t)arow * 320 + kOff;
    const float* bp = &wl[o * 320 + kOff];

    v8f c = {};
    for (int ks = 0; ks < 80; ++ks) {        // K = 320 = 80 * 4 exactly
        const v2f a = *(const v2f*)(ap + ks * 4);   // global_load_b64
        const v2f b = *(const v2f*)(bp + ks * 4);   // ds_load_b64
        c = __builtin_amdgcn_wmma_f32_16x16x4_f32(false, a, false, b,
                                                  (short)0, c, false, false);
    }

    // C fragment: vgpr r, lane -> image row m0 + r + (lane>=16 ? 8 : 0), class o.
    const bool ov   = (o < 10);
    const int rbase = m0 + ((lane >> 4) << 3);
    #pragma unroll
    for (int r = 0; r < 8; ++r) {
        const int row = rbase + r;
        const float v = fmaxf(c[r], 0.f);            // relu(logit)
        float mv = ov ? v : -1e30f;
        #pragma unroll
        for (int s = 1; s < 16; s <<= 1)             // max over 16-lane half
            mv = fmaxf(mv, __shfl_xor(mv, s, 32));
        float e = ov ? __expf(v - mv) : 0.f;
        float sum = e;
        #pragma unroll
        for (int s = 1; s < 16; s <<= 1)             // sum over 16-lane half
            sum += __shfl_xor(sum, s, 32);
        if (ov) {
            float acc = 0.f;
            #pragma unroll
            for (int k = 0; k < NKS; ++k)            // + sum_k delta_6
                acc += d6[k * 10240 + row * 10 + o];
            out[row * 10 + o] = e / sum + acc;
        }
    }
}

extern "C" void kernel_launch(void* const* d_in, const int* in_sizes, int n_in,
                              void* d_out, int out_size, void* d_ws, size_t ws_size,
                              hipStream_t stream) {
    (void)in_sizes; (void)n_in; (void)out_size; (void)ws_size;
    const float* x     = (const float*)d_in[0];   // [1024,1,28,28]
    const float* dinit = (const float*)d_in[1];   // [16,1024,1,28,28]
    const float* W1    = (const float*)d_in[2];   // [10,1,5,5]
    const float* W2    = (const float*)d_in[3];   // [20,10,5,5]
    const float* Wfc   = (const float*)d_in[4];   // [10,320]
    const float* d6    = (const float*)d_in[11];  // [16,1024,10]
    // delta_1..delta_5 cancel algebraically and are never read.
    float* out = (float*)d_out;                   // [1024,10]
    float* a2  = (float*)d_ws;                    // [1024,320] scratch

    fused_conv_kernel<<<1024, 256, 0, stream>>>(x, dinit, W1, W2, a2);
    fc_softmax_kernel<<<8, 256, 0, stream>>>(a2, Wfc, d6, out);
}